// ExplicitFlowMatcher_15066745274532
// MI455X (gfx1250) — compile-verified
//
#include <hip/hip_runtime.h>
#include <math.h>

typedef __attribute__((ext_vector_type(2))) float v2f;
typedef __attribute__((ext_vector_type(4))) float v4f;
typedef __attribute__((ext_vector_type(8))) float v8f;

#define F_ETA 1e-5f
#define F_MIN_SIGMA 1e-6f

// ---------------------------------------------------------------------------
// K1: per-row norms of X (N x 128) and dots against all 32 xt rows.
// One wave per row; wave32 shuffle reduction.
// ---------------------------------------------------------------------------
__global__ __launch_bounds__(32) void row_norms_dots(const float* __restrict__ X,
                                                     const float* __restrict__ xt,
                                                     float* __restrict__ xn,
                                                     float* __restrict__ xtX)
{
    int row  = blockIdx.x;
    int lane = threadIdx.x;
    const float* r = X + (size_t)row * 128;
    float v0 = r[lane], v1 = r[lane + 32], v2 = r[lane + 64], v3 = r[lane + 96];
    float s = v0 * v0 + v1 * v1 + v2 * v2 + v3 * v3;
    for (int off = 16; off > 0; off >>= 1) s += __shfl_xor(s, off, 32);
    if (lane == 0) xn[row] = s;
    for (int b = 0; b < 32; b++) {
        const float* q = xt + (size_t)b * 128;
        float d = v0 * q[lane] + v1 * q[lane + 32] + v2 * q[lane + 64] + v3 * q[lane + 96];
        for (int off = 16; off > 0; off >>= 1) d += __shfl_xor(d, off, 32);
        if (lane == 0) xtX[(size_t)b * 1024 + row] = d;
    }
}

// ---------------------------------------------------------------------------
// K2: ||xt_b||^2 for the 32 query rows.
// ---------------------------------------------------------------------------
__global__ __launch_bounds__(32) void xt_norms(const float* __restrict__ xt,
                                               float* __restrict__ xtn)
{
    int b = blockIdx.x, lane = threadIdx.x;
    const float* r = xt + (size_t)b * 128;
    float v0 = r[lane], v1 = r[lane + 32], v2 = r[lane + 64], v3 = r[lane + 96];
    float s = v0 * v0 + v1 * v1 + v2 * v2 + v3 * v3;
    for (int off = 16; off > 0; off >>= 1) s += __shfl_xor(s, off, 32);
    if (lane == 0) xtn[b] = s;
}

// ---------------------------------------------------------------------------
// K3: per-batch scalar constants.
// bc[b*8+..] = {t, 1-t, invB=1/(2*bridge^2), Lb=128*log(bridge),
//               kb=2*t*(1-t), c_t, a, b}
// ---------------------------------------------------------------------------
__global__ void bconst_kernel(const float* __restrict__ t, float* __restrict__ bc)
{
    int b = threadIdx.x;
    if (b >= 32) return;
    float tb  = t[b];
    float omt = 1.0f - tb;
    float br2 = (tb + F_ETA) * (1.0f - tb + F_ETA);   // bridge^2
    float bridge = sqrtf(br2);
    float invB = 1.0f / (2.0f * br2);
    float Lb   = 128.0f * logf(bridge);
    float kb   = 2.0f * tb * omt;
    float ct   = (1.0f - 2.0f * tb) / (2.0f * br2);
    float a_   = -1.0f - ct * omt;
    float bb   =  1.0f - ct * tb;
    bc[b * 8 + 0] = tb;
    bc[b * 8 + 1] = omt;
    bc[b * 8 + 2] = invB;
    bc[b * 8 + 3] = Lb;
    bc[b * 8 + 4] = kb;
    bc[b * 8 + 5] = ct;
    bc[b * 8 + 6] = a_;
    bc[b * 8 + 7] = bb;
}

// ---------------------------------------------------------------------------
// K4: dot01 = x0 @ x1^T via V_WMMA_F32_16X16X4_F32 with 2x2 register blocking:
// one wave computes a 32x32 output region (4 accumulators), so each A/B frag
// feeds two WMMAs and every source row is fetched half as often from L2.
// Fused epilogue produces dotm=dot01, invm=1/sigma^2, lm=128*log(sigma).
//
// Frag layouts (fp32, wave32):
//  A 16x4:  lane l<16 -> row M=l, {K=k,k+1}; lanes 16-31 -> row M=l-16,
//           {K=k+2,k+3}  => contiguous float2 at row + k + 2*half.
//  B 4x16:  vgpr0 = K=k (lanes 0-15) / K=k+1 (lanes 16-31); vgpr1 = K=k+2 /
//           K=k+3; N = lane%16 => elements {k+half, k+2+half} of x1 row,
//           extracted from one b128 load with two selects.
//  C/D 16x16: vgpr r -> M = r + 8*half, N = lane%16.
// ---------------------------------------------------------------------------
__global__ __launch_bounds__(32) void dot01_wmma(const float* __restrict__ x0,
                                                 const float* __restrict__ x1,
                                                 const float* __restrict__ x0n,
                                                 const float* __restrict__ x1n,
                                                 const float* __restrict__ sigw,
                                                 float* __restrict__ dotm,
                                                 float* __restrict__ invm,
                                                 float* __restrict__ lm)
{
    int i0 = blockIdx.x * 32, j0 = blockIdx.y * 32;
    int lane = threadIdx.x;
    int half = lane >> 4;      // 0 or 1
    int l16  = lane & 15;
    const float* ar0 = x0 + (size_t)(i0 + l16) * 128;
    const float* ar1 = x0 + (size_t)(i0 + 16 + l16) * 128;
    const float* br0 = x1 + (size_t)(j0 + l16) * 128;
    const float* br1 = x1 + (size_t)(j0 + 16 + l16) * 128;

    v8f acc00 = {}, acc01 = {}, acc10 = {}, acc11 = {};
    for (int k = 0; k < 128; k += 4) {
        v2f a0 = *(const v2f*)(ar0 + k + 2 * half);
        v2f a1 = *(const v2f*)(ar1 + k + 2 * half);
        v4f b0v = *(const v4f*)(br0 + k);
        v4f b1v = *(const v4f*)(br1 + k);
        v2f b0, b1;
        b0.x = half ? b0v.y : b0v.x;
        b0.y = half ? b0v.w : b0v.z;
        b1.x = half ? b1v.y : b1v.x;
        b1.y = half ? b1v.w : b1v.z;
        acc00 = __builtin_amdgcn_wmma_f32_16x16x4_f32(false, a0, false, b0,
                                                      (short)0, acc00, false, false);
        acc01 = __builtin_amdgcn_wmma_f32_16x16x4_f32(false, a0, false, b1,
                                                      (short)0, acc01, false, false);
        acc10 = __builtin_amdgcn_wmma_f32_16x16x4_f32(false, a1, false, b0,
                                                      (short)0, acc10, false, false);
        acc11 = __builtin_amdgcn_wmma_f32_16x16x4_f32(false, a1, false, b1,
                                                      (short)0, acc11, false, false);
    }

    // softplus(sigma_w[0]), numerically stable
    float sw = sigw[0];
    float sp = fmaxf(sw, 0.0f) + log1pf(expf(-fabsf(sw)));

    v8f accs[4] = {acc00, acc01, acc10, acc11};
    for (int mi = 0; mi < 2; mi++) {
        for (int nj = 0; nj < 2; nj++) {
            v8f acc = accs[mi * 2 + nj];
            for (int r = 0; r < 8; r++) {
                int i = i0 + mi * 16 + r + 8 * half;
                int j = j0 + nj * 16 + l16;
                float d    = acc[r];
                float dist = x0n[i] + x1n[j] - 2.0f * d;
                float s    = sp * sqrtf(fmaxf(dist, 0.0f) * (1.0f / 128.0f)) + F_MIN_SIGMA;
                size_t g = (size_t)i * 1024 + j;
                dotm[g] = d;
                invm[g] = 1.0f / (s * s);
                lm[g]   = 128.0f * logf(s);
            }
        }
    }
}

// ---------------------------------------------------------------------------
// Tiled log-weight passes. Tile = 64 rows (i) x 32 cols (j), grid (16,32).
// logw = -(xtn[b] + p[b,i] + q[b,j] + kb*dot)*invB*inv - L - Lb
//   p[b,i] = (1-t)^2*x0n[i] - 2(1-t)*xt_x0[b,i]
//   q[b,j] =  t^2  *x1n[j] - 2 t   *xt_x1[b,j]
// ---------------------------------------------------------------------------
__global__ __launch_bounds__(256) void pass_max(const float* __restrict__ dotm,
                                                const float* __restrict__ invm,
                                                const float* __restrict__ lm,
                                                const float* __restrict__ x0n,
                                                const float* __restrict__ x1n,
                                                const float* __restrict__ xtn,
                                                const float* __restrict__ xtx0,
                                                const float* __restrict__ xtx1,
                                                const float* __restrict__ bc,
                                                float* __restrict__ pmax)
{
    __shared__ float sd[2048], si[2048], sl[2048];
    __shared__ float sp_[64], sq_[32], sred[256];
    int it = blockIdx.x, jt = blockIdx.y;
    int i0 = it * 64, j0 = jt * 32;
    int t = threadIdx.x;
    for (int r = 0; r < 8; r++) {
        int pi = r * 256 + t;
        int li = pi >> 5, lj = pi & 31;
        size_t g = (size_t)(i0 + li) * 1024 + (j0 + lj);
        sd[pi] = dotm[g]; si[pi] = invm[g]; sl[pi] = lm[g];
    }
    __syncthreads();
    int blk = jt * 16 + it;   // 0..511
    for (int b = 0; b < 32; b++) {
        float tb   = bc[b * 8 + 0];
        float omt  = bc[b * 8 + 1];
        float invB = bc[b * 8 + 2];
        float Lb   = bc[b * 8 + 3];
        float kb   = bc[b * 8 + 4];
        float xb   = xtn[b];
        if (t < 64) {
            sp_[t] = omt * omt * x0n[i0 + t] - 2.0f * omt * xtx0[(size_t)b * 1024 + i0 + t];
        } else if (t < 96) {
            int j = t - 64;
            sq_[j] = tb * tb * x1n[j0 + j] - 2.0f * tb * xtx1[(size_t)b * 1024 + j0 + j];
        }
        __syncthreads();
        float mx = -INFINITY;
        for (int r = 0; r < 8; r++) {
            int pi = r * 256 + t;
            int li = pi >> 5, lj = pi & 31;
            float val = xb + sp_[li] + sq_[lj] + kb * sd[pi];
            float lw  = -val * invB * si[pi] - sl[pi] - Lb;
            mx = fmaxf(mx, lw);
        }
        sred[t] = mx;
        __syncthreads();
        for (int s = 128; s > 0; s >>= 1) {
            if (t < s) sred[t] = fmaxf(sred[t], sred[t + s]);
            __syncthreads();
        }
        if (t == 0) pmax[b * 512 + blk] = sred[0];
        __syncthreads();
    }
}

__global__ __launch_bounds__(256) void reduce_max(const float* __restrict__ pmax,
                                                  float* __restrict__ m)
{
    __shared__ float s[256];
    int b = blockIdx.x, t = threadIdx.x;
    s[t] = fmaxf(pmax[b * 512 + t], pmax[b * 512 + 256 + t]);
    __syncthreads();
    for (int k = 128; k > 0; k >>= 1) {
        if (t < k) s[t] = fmaxf(s[t], s[t + k]);
        __syncthreads();
    }
    if (t == 0) m[b] = s[0];
}

__global__ __launch_bounds__(256) void pass_w(const float* __restrict__ dotm,
                                              const float* __restrict__ invm,
                                              const float* __restrict__ lm,
                                              const float* __restrict__ x0n,
                                              const float* __restrict__ x1n,
                                              const float* __restrict__ xtn,
                                              const float* __restrict__ xtx0,
                                              const float* __restrict__ xtx1,
                                              const float* __restrict__ bc,
                                              const float* __restrict__ m,
                                              float* __restrict__ wip,
                                              float* __restrict__ wjp,
                                              float* __restrict__ dp)
{
    __shared__ float sd[2048], si[2048], sl[2048], sw[2048];
    __shared__ float sp_[64], sq_[32], srow[64];
    int it = blockIdx.x, jt = blockIdx.y;
    int i0 = it * 64, j0 = jt * 32;
    int t = threadIdx.x;
    for (int r = 0; r < 8; r++) {
        int pi = r * 256 + t;
        int li = pi >> 5, lj = pi & 31;
        size_t g = (size_t)(i0 + li) * 1024 + (j0 + lj);
        sd[pi] = dotm[g]; si[pi] = invm[g]; sl[pi] = lm[g];
    }
    __syncthreads();
    int blk = jt * 16 + it;
    for (int b = 0; b < 32; b++) {
        float tb   = bc[b * 8 + 0];
        float omt  = bc[b * 8 + 1];
        float invB = bc[b * 8 + 2];
        float Lb   = bc[b * 8 + 3];
        float kb   = bc[b * 8 + 4];
        float xb   = xtn[b];
        float mb   = m[b];
        if (t < 64) {
            sp_[t] = omt * omt * x0n[i0 + t] - 2.0f * omt * xtx0[(size_t)b * 1024 + i0 + t];
        } else if (t < 96) {
            int j = t - 64;
            sq_[j] = tb * tb * x1n[j0 + j] - 2.0f * tb * xtx1[(size_t)b * 1024 + j0 + j];
        }
        __syncthreads();
        for (int r = 0; r < 8; r++) {
            int pi = r * 256 + t;
            int li = pi >> 5, lj = pi & 31;
            float val = xb + sp_[li] + sq_[lj] + kb * sd[pi];
            float lw  = -val * invB * si[pi] - sl[pi] - Lb;
            sw[pi] = expf(lw - mb);
        }
        __syncthreads();
        if (t < 64) {                       // row sums -> wi partial over jt
            float s = 0.0f;
            for (int j = 0; j < 32; j++) s += sw[t * 32 + j];
            srow[t] = s;
            wip[((size_t)(b * 32 + jt)) * 1024 + (i0 + t)] = s;
        } else if (t < 96) {                // column sums -> wj partial over it
            int lj = t - 64;
            float s = 0.0f;
            for (int i = 0; i < 64; i++) s += sw[i * 32 + lj];
            wjp[((size_t)(b * 16 + it)) * 1024 + (j0 + lj)] = s;
        }
        __syncthreads();
        if (t == 0) {
            float s = 0.0f;
            for (int i = 0; i < 64; i++) s += srow[i];
            dp[b * 512 + blk] = s;
        }
        __syncthreads();
    }
}

__global__ __launch_bounds__(1024) void reduce_wi(const float* __restrict__ wip,
                                                  float* __restrict__ wi)
{
    int b = blockIdx.x, i = threadIdx.x;
    float s = 0.0f;
    for (int jt = 0; jt < 32; jt++) s += wip[((size_t)(b * 32 + jt)) * 1024 + i];
    wi[(size_t)b * 1024 + i] = s;
}

__global__ __launch_bounds__(1024) void reduce_wj(const float* __restrict__ wjp,
                                                  float* __restrict__ wj)
{
    int b = blockIdx.x, j = threadIdx.x;
    float s = 0.0f;
    for (int it = 0; it < 16; it++) s += wjp[((size_t)(b * 16 + it)) * 1024 + j];
    wj[(size_t)b * 1024 + j] = s;
}

__global__ __launch_bounds__(256) void reduce_den(const float* __restrict__ dp,
                                                  float* __restrict__ den)
{
    __shared__ float s[256];
    int b = blockIdx.x, t = threadIdx.x;
    s[t] = dp[b * 512 + t] + dp[b * 512 + 256 + t];
    __syncthreads();
    for (int k = 128; k > 0; k >>= 1) {
        if (t < k) s[t] += s[t + k];
        __syncthreads();
    }
    if (t == 0) den[b] = s[0];
}

// ---------------------------------------------------------------------------
// K5: fused output GEMV: out[b,d] = (ct*xt*den + a*wi.x0 + b*wj.x1)/max(den,1e-12)
// ---------------------------------------------------------------------------
__global__ __launch_bounds__(128) void final_out(const float* __restrict__ xt,
                                                 const float* __restrict__ x0,
                                                 const float* __restrict__ x1,
                                                 const float* __restrict__ wi,
                                                 const float* __restrict__ wj,
                                                 const float* __restrict__ den,
                                                 const float* __restrict__ bc,
                                                 float* __restrict__ out)
{
    int b = blockIdx.x, d = threadIdx.x;
    float ct = bc[b * 8 + 5], a_ = bc[b * 8 + 6], bb = bc[b * 8 + 7];
    float wx0 = 0.0f, wx1 = 0.0f;
    for (int i = 0; i < 1024; i++)
        wx0 = fmaf(wi[(size_t)b * 1024 + i], x0[(size_t)i * 128 + d], wx0);
    for (int j = 0; j < 1024; j++)
        wx1 = fmaf(wj[(size_t)b * 1024 + j], x1[(size_t)j * 128 + d], wx1);
    float dn = den[b];
    float num = ct * xt[(size_t)b * 128 + d] * dn + a_ * wx0 + bb * wx1;
    out[(size_t)b * 128 + d] = num / fmaxf(dn, 1e-12f);
}

// ---------------------------------------------------------------------------
extern "C" void kernel_launch(void* const* d_in, const int* in_sizes, int n_in,
                              void* d_out, int out_size, void* d_ws, size_t ws_size,
                              hipStream_t stream)
{
    const float* xt   = (const float*)d_in[0];   // 32 x 128
    const float* t    = (const float*)d_in[1];   // 32
    const float* x0   = (const float*)d_in[2];   // 1024 x 128
    const float* x1   = (const float*)d_in[3];   // 1024 x 128
    const float* sigw = (const float*)d_in[4];   // 1
    float* out = (float*)d_out;                  // 32 x 128

    float* ws = (float*)d_ws;
    size_t cur = 0;
    auto alloc = [&](size_t n) { float* p = ws + cur; cur += n; return p; };

    float* x0n  = alloc(1024);
    float* x1n  = alloc(1024);
    float* xtn  = alloc(32);
    float* bc   = alloc(32 * 8);
    float* xtx0 = alloc(32 * 1024);
    float* xtx1 = alloc(32 * 1024);
    float* dotm = alloc(1024 * 1024);
    float* invm = alloc(1024 * 1024);
    float* lm   = alloc(1024 * 1024);
    float* pmax = alloc(32 * 512);
    float* m    = alloc(32);
    float* dp   = alloc(32 * 512);
    float* den  = alloc(32);
    float* wip  = alloc((size_t)32 * 32 * 1024);
    float* wjp  = alloc((size_t)32 * 16 * 1024);
    float* wi   = alloc(32 * 1024);
    float* wj   = alloc(32 * 1024);
    (void)ws_size; (void)in_sizes; (void)n_in; (void)out_size;

    row_norms_dots<<<1024, 32, 0, stream>>>(x0, xt, x0n, xtx0);
    row_norms_dots<<<1024, 32, 0, stream>>>(x1, xt, x1n, xtx1);
    xt_norms<<<32, 32, 0, stream>>>(xt, xtn);
    bconst_kernel<<<1, 32, 0, stream>>>(t, bc);

    dot01_wmma<<<dim3(32, 32), 32, 0, stream>>>(x0, x1, x0n, x1n, sigw, dotm, invm, lm);

    pass_max<<<dim3(16, 32), 256, 0, stream>>>(dotm, invm, lm, x0n, x1n, xtn,
                                               xtx0, xtx1, bc, pmax);
    reduce_max<<<32, 256, 0, stream>>>(pmax, m);

    pass_w<<<dim3(16, 32), 256, 0, stream>>>(dotm, invm, lm, x0n, x1n, xtn,
                                             xtx0, xtx1, bc, m, wip, wjp, dp);
    reduce_wi<<<32, 1024, 0, stream>>>(wip, wi);
    reduce_wj<<<32, 1024, 0, stream>>>(wjp, wj);
    reduce_den<<<32, 256, 0, stream>>>(dp, den);

    final_out<<<32, 128, 0, stream>>>(xt, x0, x1, wi, wj, den, bc, out);
}